// MultiheadAttentionMechanism_47828755808484
// MI455X (gfx1250) — compile-verified
//
#include <hip/hip_runtime.h>
#include <hip/hip_bf16.h>

typedef __attribute__((ext_vector_type(2))) float v2f;
typedef __attribute__((ext_vector_type(8))) float v8f;

#define B_   32
#define L_   2048
#define KD_  512
#define HD_  512                    // H*D
#define NCHUNK 16
#define ROWS_PER_CHUNK (L_ / NCHUNK)  // 128

// ---------------------------------------------------------------------------
// Phase 1: partial column sums of value over L.
// grid = (B, NCHUNK), block = 128 threads; thread t owns columns 4t..4t+3
// (float4 loads -> global_load_b128, fully coalesced: 2KB per row per block).
// ---------------------------------------------------------------------------
__global__ void __launch_bounds__(128)
colsum_partial_kernel(const float* __restrict__ value, float* __restrict__ partial) {
    const int b     = blockIdx.x;
    const int chunk = blockIdx.y;
    const int col   = threadIdx.x * 4;

    const float* p = value + ((size_t)b * L_ + (size_t)chunk * ROWS_PER_CHUNK) * KD_ + col;
    float4 acc = make_float4(0.f, 0.f, 0.f, 0.f);
    for (int l = 0; l < ROWS_PER_CHUNK; ++l) {
        __builtin_prefetch(p + 8 * KD_, 0, 0);   // global_prefetch_b8, stream ahead
        float4 v = *(const float4*)p;
        acc.x += v.x; acc.y += v.y; acc.z += v.z; acc.w += v.w;
        p += KD_;
    }
    *(float4*)(partial + (size_t)(b * NCHUNK + chunk) * KD_ + col) = acc;
}

// ---------------------------------------------------------------------------
// Phase 2: deterministic reduce of the NCHUNK partials -> S[B, KD].
// ---------------------------------------------------------------------------
__global__ void __launch_bounds__(256)
colsum_reduce_kernel(const float* __restrict__ partial, float* __restrict__ S) {
    const int i  = blockIdx.x * blockDim.x + threadIdx.x;   // < B_*KD_
    const int b  = i / KD_;
    const int kd = i - b * KD_;
    float acc = 0.f;
    for (int c = 0; c < NCHUNK; ++c)
        acc += partial[(size_t)(b * NCHUNK + c) * KD_ + kd];
    S[i] = acc;
}

// ---------------------------------------------------------------------------
// Phase 3: f32 WMMA GEMM  Out[M,N] = A[M,K] @ Bm[K,N] (+ bias).
// One wave per 16x16 output tile, K stepped by 4 via V_WMMA_F32_16X16X4_F32.
// A-frag (16x4 f32, ISA 7.12.2): lane half selects K-pair; M = lane&15.
// B-frag (4x16): VGPR0 holds K={0|2}, VGPR1 K={1|3}; N = lane&15.
// C/D (16x16 f32): VGPR i -> row i + 8*(lane>>4), col = lane&15.
// ---------------------------------------------------------------------------
__global__ void __launch_bounds__(128)
wmma_gemm_f32_kernel(const float* __restrict__ A, const float* __restrict__ Bm,
                     const float* __restrict__ bias, float* __restrict__ Out,
                     int M, int N, int K) {
    const int wave   = blockIdx.x * (blockDim.x >> 5) + (threadIdx.x >> 5);
    const int tilesN = N >> 4;
    const int tm     = wave / tilesN;
    const int tn     = wave - tm * tilesN;
    const int lane   = threadIdx.x & 31;
    const int half   = lane >> 4;
    const int mrow   = lane & 15;

    const float* arow = A  + (size_t)(tm * 16 + mrow) * K;
    const float* bcol = Bm + (tn * 16 + mrow);

    v8f c = {};
    for (int k0 = 0; k0 < K; k0 += 4) {
        const int kk = k0 + half * 2;
        v2f a, b;
        a.x = arow[kk];
        a.y = arow[kk + 1];
        b.x = bcol[(size_t)kk * N];
        b.y = bcol[(size_t)(kk + 1) * N];
        c = __builtin_amdgcn_wmma_f32_16x16x4_f32(
                /*neg_a=*/false, a, /*neg_b=*/false, b,
                /*c_mod=*/(short)0, c, /*reuse_a=*/false, /*reuse_b=*/false);
    }

    const int   col = tn * 16 + mrow;
    const float bv  = bias ? bias[col] : 0.0f;
#pragma unroll
    for (int i = 0; i < 8; ++i) {
        const int row = tm * 16 + i + half * 8;
        Out[(size_t)row * N + col] = c[i] + bv;
    }
}

// ---------------------------------------------------------------------------
// Phase 4: aw = softmax over a singleton axis == all ones.
// ---------------------------------------------------------------------------
__global__ void __launch_bounds__(256)
fill_ones_kernel(float* __restrict__ out, int n) {
    const int i = blockIdx.x * blockDim.x + threadIdx.x;
    if (i < n) out[i] = 1.0f;
}

// ---------------------------------------------------------------------------
// inputs: 0=key 1=value 2=query 3=key_lens 4=Wk 5=Wv 6=Wq 7=Wo 8=bo
// d_out:  cv [32*512] floats, then aw [4*32*2048] floats (all ones).
// ---------------------------------------------------------------------------
extern "C" void kernel_launch(void* const* d_in, const int* in_sizes, int n_in,
                              void* d_out, int out_size, void* d_ws, size_t ws_size,
                              hipStream_t stream) {
    (void)in_sizes; (void)n_in; (void)out_size; (void)ws_size;
    const float* value = (const float*)d_in[1];
    const float* Wv    = (const float*)d_in[5];
    const float* Wo    = (const float*)d_in[7];
    const float* bo    = (const float*)d_in[8];
    float*       out   = (float*)d_out;

    float* ws      = (float*)d_ws;
    float* partial = ws;                          // B*NCHUNK*KD = 262144 floats
    float* S       = partial + B_ * NCHUNK * KD_; // B*KD  = 16384 floats
    float* T       = S + B_ * KD_;                // B*HD  = 16384 floats

    dim3 g1(B_, NCHUNK);
    colsum_partial_kernel<<<g1, 128, 0, stream>>>(value, partial);
    colsum_reduce_kernel<<<(B_ * KD_) / 256, 256, 0, stream>>>(partial, S);

    // 64 tiles (2 x 32), 4 waves/block -> 16 blocks exactly cover the grid.
    wmma_gemm_f32_kernel<<<16, 128, 0, stream>>>(S, Wv, nullptr, T, B_, HD_, KD_);
    wmma_gemm_f32_kernel<<<16, 128, 0, stream>>>(T, Wo, bo, out, B_, KD_, HD_);

    const int aw_n = 4 * B_ * L_;   // H*B*L = 262144
    fill_ones_kernel<<<aw_n / 256, 256, 0, stream>>>(out + B_ * KD_, aw_n);
}